// TextGCN_71614284694307
// MI455X (gfx1250) — compile-verified
//
#include <hip/hip_runtime.h>
#include <hip/hip_bf16.h>

typedef __attribute__((ext_vector_type(2))) float v2f;
typedef __attribute__((ext_vector_type(8))) float v8f;

#define HIDDEN 128
#define NCLASS 16

// ---------------- gcn_norm ----------------

__global__ void k_init_deg(float* __restrict__ deg, int n) {
  int i = blockIdx.x * blockDim.x + threadIdx.x;
  if (i < n) deg[i] = 1.0f;  // self-loop weight
}

__global__ void k_accum_deg(const int* __restrict__ col, const float* __restrict__ w,
                            float* __restrict__ deg, int E) {
  int e = blockIdx.x * blockDim.x + threadIdx.x;
  if (e < E) atomicAdd(&deg[col[e]], w[e]);
}

__global__ void k_dinv(float* __restrict__ deg, int n) {
  int i = blockIdx.x * blockDim.x + threadIdx.x;
  if (i < n) {
    float d = deg[i];
    deg[i] = (d > 0.0f) ? rsqrtf(d) : 0.0f;   // deg buffer becomes dinv
  }
}

// ---------------- GEMM1: H = X[N,128] @ W[128,128] via V_WMMA_F32_16X16X4_F32 ----------------
// One block = one 16-row M tile (A staged in LDS); 8 waves cover N=128 in 16-wide tiles.

__global__ void __launch_bounds__(256) k_gemm1(const float* __restrict__ X,
                                               const float* __restrict__ W,
                                               float* __restrict__ H, int n) {
  __shared__ float As[16 * HIDDEN];  // 8 KB
  const int m0 = blockIdx.x * 16;
  const int tid = threadIdx.x;

  // cooperative tile load (rows clamped for tail blocks; N%16==0 here anyway)
  for (int idx = tid; idx < 16 * HIDDEN / 4; idx += 256) {
    int rrow = idx / (HIDDEN / 4);
    int rcol = idx % (HIDDEN / 4);
    int grow = m0 + rrow; if (grow > n - 1) grow = n - 1;
    ((float4*)As)[idx] = ((const float4*)(X + (size_t)grow * HIDDEN))[rcol];
  }
  __syncthreads();

  const int wave = tid >> 5;
  const int lane = tid & 31;
  const int n0 = wave * 16;
  const int mA = lane & 15;          // A-frag: M = lane%16
  const int kg = (lane >> 4) << 1;   // A-frag: lanes 16-31 hold K=2,3
  const int nB = lane & 15;          // B-frag: N = lane%16 (mirrored layout)

  v8f acc = {};
  for (int k0 = 0; k0 < HIDDEN; k0 += 4) {
    v2f a, b;
    a.x = As[mA * HIDDEN + k0 + kg];
    a.y = As[mA * HIDDEN + k0 + kg + 1];
    b.x = W[(k0 + kg) * HIDDEN + n0 + nB];
    b.y = W[(k0 + kg + 1) * HIDDEN + n0 + nB];
    acc = __builtin_amdgcn_wmma_f32_16x16x4_f32(false, a, false, b,
                                                (short)0, acc, false, false);
  }

  // C/D layout: VGPR r -> M = (lane/16)*8 + r, N = lane%16
  const int nc = n0 + (lane & 15);
  const int mBase = (lane >> 4) * 8;
  #pragma unroll
  for (int r = 0; r < 8; ++r) {
    int m = m0 + mBase + r;
    if (m < n) H[(size_t)m * HIDDEN + nc] = acc[r];
  }
}

// ---------------- GEMM2: H = X[N,128] @ W[128,16]; one wave per 16x16 tile ----------------

__global__ void __launch_bounds__(256) k_gemm2(const float* __restrict__ X,
                                               const float* __restrict__ W,
                                               float* __restrict__ H, int n) {
  const int wave = threadIdx.x >> 5;
  const int lane = threadIdx.x & 31;
  const int m0 = (blockIdx.x * 8 + wave) * 16;
  if (m0 >= n) return;  // wave-uniform: EXEC stays all-ones for WMMA

  const int mA = lane & 15;
  const int kg = (lane >> 4) << 1;
  const int nB = lane & 15;

  v8f acc = {};
  for (int k0 = 0; k0 < HIDDEN; k0 += 4) {
    int mra = m0 + mA; if (mra > n - 1) mra = n - 1;
    v2f a, b;
    a.x = X[(size_t)mra * HIDDEN + k0 + kg];
    a.y = X[(size_t)mra * HIDDEN + k0 + kg + 1];
    b.x = W[(k0 + kg) * NCLASS + nB];
    b.y = W[(k0 + kg + 1) * NCLASS + nB];
    acc = __builtin_amdgcn_wmma_f32_16x16x4_f32(false, a, false, b,
                                                (short)0, acc, false, false);
  }

  const int nc = lane & 15;
  const int mBase = (lane >> 4) * 8;
  #pragma unroll
  for (int r = 0; r < 8; ++r) {
    int m = m0 + mBase + r;
    if (m < n) H[(size_t)m * NCLASS + nc] = acc[r];
  }
}

// ---------------- aggregation ----------------

// agg[i,f] = h[i,f] * dinv[i]^2   (self-loop term; also initializes the buffer)
__global__ void k_selfloop(const float* __restrict__ h, const float* __restrict__ dinv,
                           float* __restrict__ agg, int n, int F) {
  int idx = blockIdx.x * blockDim.x + threadIdx.x;
  if (idx < n * F) {
    int i = idx / F;
    float di = dinv[i];
    agg[idx] = h[idx] * di * di;
  }
}

// one wave per edge; lane handles a float4 of the 128-feature row
__global__ void __launch_bounds__(256) k_edge_agg128(
    const int* __restrict__ row, const int* __restrict__ col,
    const float* __restrict__ w, const float* __restrict__ dinv,
    const float* __restrict__ h, float* __restrict__ agg, int E) {
  int e = blockIdx.x * 8 + (threadIdx.x >> 5);
  if (e >= E) return;
  int lane = threadIdx.x & 31;
  int r = row[e], c = col[e];
  float nrm = dinv[r] * w[e] * dinv[c];
  float4 hv = ((const float4*)(h + (size_t)r * HIDDEN))[lane];
  float* dst = agg + (size_t)c * HIDDEN + lane * 4;
  atomicAdd(dst + 0, hv.x * nrm);
  atomicAdd(dst + 1, hv.y * nrm);
  atomicAdd(dst + 2, hv.z * nrm);
  atomicAdd(dst + 3, hv.w * nrm);
}

// one thread per (edge, feature) for the 16-wide second layer
__global__ void k_edge_agg16(const int* __restrict__ row, const int* __restrict__ col,
                             const float* __restrict__ w, const float* __restrict__ dinv,
                             const float* __restrict__ h, float* __restrict__ agg, int E) {
  long long idx = (long long)blockIdx.x * blockDim.x + threadIdx.x;
  if (idx >= (long long)E * NCLASS) return;
  int e = (int)(idx >> 4);
  int f = (int)(idx & 15);
  int r = row[e], c = col[e];
  float nrm = dinv[r] * w[e] * dinv[c];
  atomicAdd(&agg[(size_t)c * NCLASS + f], h[(size_t)r * NCLASS + f] * nrm);
}

__global__ void k_bias_relu(float* __restrict__ x, const float* __restrict__ b, int n) {
  int idx = blockIdx.x * blockDim.x + threadIdx.x;
  if (idx < n * HIDDEN) {
    float v = x[idx] + b[idx % HIDDEN];
    x[idx] = v > 0.0f ? v : 0.0f;
  }
}

__global__ void k_bias16(float* __restrict__ x, const float* __restrict__ b, int n) {
  int idx = blockIdx.x * blockDim.x + threadIdx.x;
  if (idx < n * NCLASS) x[idx] += b[idx % NCLASS];
}

// ---------------- launch ----------------

extern "C" void kernel_launch(void* const* d_in, const int* in_sizes, int n_in,
                              void* d_out, int out_size, void* d_ws, size_t ws_size,
                              hipStream_t stream) {
  const int*   ei  = (const int*)d_in[0];      // edge_index [2, E]
  const float* ew  = (const float*)d_in[1];    // edge_weight [E]
  const float* emb = (const float*)d_in[2];    // [N, 128]
  const float* W1  = (const float*)d_in[3];    // [128, 128]
  const float* b1  = (const float*)d_in[4];    // [128]
  const float* W2  = (const float*)d_in[5];    // [128, 16]
  const float* b2  = (const float*)d_in[6];    // [16]
  float* out = (float*)d_out;                  // [N, 16]

  const int E = in_sizes[1];
  const int N = in_sizes[2] / HIDDEN;
  const int* row = ei;
  const int* col = ei + E;

  // workspace: deg/dinv (N) | h (N*128, reused as h2) | x1 (N*128)
  float* deg = (float*)d_ws;
  float* h   = deg + N;
  float* x1  = h + (size_t)N * HIDDEN;

  const int B = 256;

  // gcn_norm
  k_init_deg <<<(N + B - 1) / B, B, 0, stream>>>(deg, N);
  k_accum_deg<<<(E + B - 1) / B, B, 0, stream>>>(col, ew, deg, E);
  k_dinv     <<<(N + B - 1) / B, B, 0, stream>>>(deg, N);

  // layer 1
  k_gemm1<<<(N + 15) / 16, 256, 0, stream>>>(emb, W1, h, N);
  {
    long long tot = (long long)N * HIDDEN;
    k_selfloop<<<(unsigned)((tot + B - 1) / B), B, 0, stream>>>(h, deg, x1, N, HIDDEN);
  }
  k_edge_agg128<<<(E + 7) / 8, 256, 0, stream>>>(row, col, ew, deg, h, x1, E);
  {
    long long tot = (long long)N * HIDDEN;
    k_bias_relu<<<(unsigned)((tot + B - 1) / B), B, 0, stream>>>(x1, b1, N);
  }

  // layer 2 (h buffer reused for h2 = x1 @ W2)
  k_gemm2<<<(N + 16 * 8 - 1) / (16 * 8), 256, 0, stream>>>(x1, W2, h, N);
  {
    long long tot = (long long)N * NCLASS;
    k_selfloop<<<(unsigned)((tot + B - 1) / B), B, 0, stream>>>(h, deg, out, N, NCLASS);
  }
  {
    long long tot = (long long)E * NCLASS;
    k_edge_agg16<<<(unsigned)((tot + B - 1) / B), B, 0, stream>>>(row, col, ew, deg, h, out, E);
  }
  {
    long long tot = (long long)N * NCLASS;
    k_bias16<<<(unsigned)((tot + B - 1) / B), B, 0, stream>>>(out, b2, N);
  }
}